// Classifier_61581241090170
// MI455X (gfx1250) — compile-verified
//
#include <hip/hip_runtime.h>
#include <stdint.h>

#define DF      64
#define NGRAPH  32
#define NCLASS  40

typedef float v2f __attribute__((ext_vector_type(2)));
typedef float v8f __attribute__((ext_vector_type(8)));

// ---------- monotonic float<->uint mapping for atomic max ----------
__device__ __forceinline__ unsigned fmap(float f) {
    unsigned b = __float_as_uint(f);
    return (b & 0x80000000u) ? ~b : (b | 0x80000000u);
}
__device__ __forceinline__ float funmap(unsigned u) {
    unsigned b = (u & 0x80000000u) ? (u & 0x7fffffffu) : ~u;
    return __uint_as_float(b);
}

// ---------- small utility kernels ----------
__global__ void zero_u32(unsigned* __restrict__ p, int n) {
    int i = blockIdx.x * 256 + threadIdx.x;
    if (i < n) p[i] = 0u;
}

__global__ void count_deg(const int* __restrict__ dst, int* __restrict__ deg, int e) {
    int i = blockIdx.x * 256 + threadIdx.x;
    if (i < e) atomicAdd(&deg[dst[i]], 1);
}

// single-block exclusive scan over N elements (N ~ 100k: 98 chunks of 1024)
__global__ void __launch_bounds__(1024)
scan_excl(const int* __restrict__ deg, int* __restrict__ offs, int n) {
    __shared__ int buf[1024];
    __shared__ int carry;
    if (threadIdx.x == 0) carry = 0;
    __syncthreads();
    for (int base = 0; base < n; base += 1024) {
        int i = base + (int)threadIdx.x;
        int v = (i < n) ? deg[i] : 0;
        buf[threadIdx.x] = v;
        __syncthreads();
        for (int s = 1; s < 1024; s <<= 1) {
            int t = (threadIdx.x >= (unsigned)s) ? buf[threadIdx.x - s] : 0;
            __syncthreads();
            buf[threadIdx.x] += t;
            __syncthreads();
        }
        int incl = buf[threadIdx.x];
        int c = carry;
        if (i < n) offs[i] = c + incl - v;
        __syncthreads();
        if (threadIdx.x == 1023) carry = c + buf[1023];
        __syncthreads();
    }
    if (threadIdx.x == 0) offs[n] = carry;
}

__global__ void fill_csr(const int* __restrict__ src, const int* __restrict__ dst,
                         const int* __restrict__ offs, int* __restrict__ cursor,
                         int* __restrict__ csr, int e) {
    int i = blockIdx.x * 256 + threadIdx.x;
    if (i >= e) return;
    int d = dst[i];
    int pos = offs[d] + atomicAdd(&cursor[d], 1);
    csr[pos] = src[i];
}

// ---------- dual GEMM via fp32 WMMA: t = h*Tw ; p = h*Pw + (tb+pb) ----------
// block = 256 threads = 8 waves; each wave computes a 16x64 strip of t and p.
__global__ void __launch_bounds__(256)
gemm_dual_wmma(const float* __restrict__ h,
               const float* __restrict__ tw, const float* __restrict__ pw,
               const float* __restrict__ tb, const float* __restrict__ pb,
               float* __restrict__ tout, float* __restrict__ pout, int n) {
    __shared__ float s_tw[DF * DF];
    __shared__ float s_pw[DF * DF];
    for (int i = threadIdx.x; i < DF * DF; i += 256) {
        s_tw[i] = tw[i];
        s_pw[i] = pw[i];
    }
    __syncthreads();

    const int lane  = threadIdx.x & 31;
    const int wave  = threadIdx.x >> 5;
    const int mbase = blockIdx.x * 128 + wave * 16;
    if (mbase >= n) return;              // wave-uniform: EXEC stays all-ones

    const int mlane = lane & 15;         // M index (A) / N index (B,C,D)
    const int khalf = (lane >> 4) * 2;   // K sub-offset 0 or 2 per ISA A/B layout
    const float* hrow = h + (size_t)(mbase + mlane) * DF;

    v8f acc_t[4] = {};
    v8f acc_p[4] = {};

    for (int ks = 0; ks < 16; ++ks) {
        const int k0 = ks * 4 + khalf;
        v2f a;
        a.x = hrow[k0];
        a.y = hrow[k0 + 1];
#pragma unroll
        for (int nt = 0; nt < 4; ++nt) {
            const int col = nt * 16 + mlane;
            v2f bt, bp;
            bt.x = s_tw[k0 * DF + col];
            bt.y = s_tw[(k0 + 1) * DF + col];
            bp.x = s_pw[k0 * DF + col];
            bp.y = s_pw[(k0 + 1) * DF + col];
            acc_t[nt] = __builtin_amdgcn_wmma_f32_16x16x4_f32(
                false, a, false, bt, (short)0, acc_t[nt], false, false);
            acc_p[nt] = __builtin_amdgcn_wmma_f32_16x16x4_f32(
                false, a, false, bp, (short)0, acc_p[nt], false, false);
        }
    }

    const int mro = (lane >> 4) * 8;     // C/D layout: upper lanes hold M+8
#pragma unroll
    for (int nt = 0; nt < 4; ++nt) {
        const int col  = nt * 16 + mlane;
        const float bi = tb[col] + pb[col];
#pragma unroll
        for (int r = 0; r < 8; ++r) {
            const size_t m = (size_t)(mbase + mro + r);
            tout[m * DF + col] = acc_t[nt][r];
            pout[m * DF + col] = acc_p[nt][r] + bi;
        }
    }
}

// ---------- per-node neighbor max; writes result in place over p ----------
// out[i,c] = (max_{j in N(i)} t[j,c]) - t[i,c] + p[i,c] ; 0 if no in-edges.
// One wave per node: 32 lanes x float2 = 64 features -> one b64 load per edge.
__global__ void __launch_bounds__(256)
edge_agg(const float* __restrict__ t, float* __restrict__ p,
         const int* __restrict__ offs, const int* __restrict__ csr, int n) {
    int node = blockIdx.x * 8 + (threadIdx.x >> 5);
    int lane = threadIdx.x & 31;
    if (node >= n) return;
    int beg = offs[node], end = offs[node + 1];

    const float2* __restrict__ t2 = (const float2*)t;
    float2*       __restrict__ p2 = (float2*)p;
    const size_t base = (size_t)node * 32 + lane;

    if (beg == end) {
        p2[base] = make_float2(0.0f, 0.0f);
        return;
    }

    float m0 = -3.402823466e+38f, m1 = -3.402823466e+38f;
    int e = beg;
    for (; e + 1 < end; e += 2) {          // 2 b64 loads in flight per iter
        int j0 = csr[e];                   // wave-uniform -> scalar loads
        int j1 = csr[e + 1];
        float2 a = t2[(size_t)j0 * 32 + lane];
        float2 b = t2[(size_t)j1 * 32 + lane];
        m0 = fmaxf(m0, fmaxf(a.x, b.x));
        m1 = fmaxf(m1, fmaxf(a.y, b.y));
    }
    if (e < end) {
        int j = csr[e];
        float2 a = t2[(size_t)j * 32 + lane];
        m0 = fmaxf(m0, a.x);
        m1 = fmaxf(m1, a.y);
    }

    float2 ti = t2[base];
    float2 pi = p2[base];
    p2[base] = make_float2(m0 - ti.x + pi.x, m1 - ti.y + pi.y);
}

// ---------- graph max pooling (graph_ids sorted -> segmented reduce) ----------
// Each 64-thread group scans 64 consecutive nodes, keeps a running max per
// feature, and only issues an atomic when the (sorted) graph id changes.
__global__ void __launch_bounds__(256)
pool_max(const float* __restrict__ h, const int* __restrict__ gid,
         unsigned* __restrict__ hg, int n) {
    int c    = threadIdx.x & 63;
    int seg  = blockIdx.x * 4 + (threadIdx.x >> 6);
    int i0   = seg * 64;
    if (i0 >= n) return;
    int i1 = i0 + 64; if (i1 > n) i1 = n;

    int   curg = gid[i0];
    float m    = -3.402823466e+38f;
    for (int i = i0; i < i1; ++i) {
        int g = gid[i];                           // uniform per wave
        if (g != curg) {
            atomicMax(&hg[curg * DF + c], fmap(m));
            m = -3.402823466e+38f;
            curg = g;
        }
        m = fmaxf(m, h[(size_t)i * DF + c]);
    }
    atomicMax(&hg[curg * DF + c], fmap(m));
}

// ---------- tiny classifier: out[g,c] = hg[g,:] . W[:,c] + b[c] ----------
__global__ void __launch_bounds__(1024)
classify(const unsigned* __restrict__ hg_u, const float* __restrict__ w,
         const float* __restrict__ b, float* __restrict__ out) {
    __shared__ float hg[NGRAPH * DF];
    for (int i = threadIdx.x; i < NGRAPH * DF; i += 1024) hg[i] = funmap(hg_u[i]);
    __syncthreads();
    for (int idx = threadIdx.x; idx < NGRAPH * NCLASS; idx += 1024) {
        int g = idx / NCLASS, cls = idx % NCLASS;
        float acc = b[cls];
#pragma unroll
        for (int k = 0; k < DF; ++k) acc += hg[g * DF + k] * w[k * NCLASS + cls];
        out[idx] = acc;
    }
}

extern "C" void kernel_launch(void* const* d_in, const int* in_sizes, int n_in,
                              void* d_out, int out_size, void* d_ws, size_t ws_size,
                              hipStream_t stream) {
    (void)n_in; (void)out_size; (void)ws_size;
    const float* h   = (const float*)d_in[0];
    const int*   src = (const int*)  d_in[1];
    const int*   dst = (const int*)  d_in[2];
    const int*   gid = (const int*)  d_in[3];
    const float* thw = (const float*)d_in[4];
    const float* thb = (const float*)d_in[5];
    const float* phw = (const float*)d_in[6];
    const float* phb = (const float*)d_in[7];
    const float* clw = (const float*)d_in[8];
    const float* clb = (const float*)d_in[9];

    const int N = in_sizes[0] / DF;
    const int E = in_sizes[1];

    char* ws = (char*)d_ws;
    size_t o = 0;
#define TAKE(ty, name, bytes) ty name = (ty)(ws + o); o = (o + (size_t)(bytes) + 255) & ~(size_t)255
    TAKE(float*,    t_buf,  (size_t)N * DF * 4);
    TAKE(float*,    p0,     (size_t)N * DF * 4);
    TAKE(float*,    p1,     (size_t)N * DF * 4);
    TAKE(int*,      csr,    (size_t)E * 4);
    TAKE(int*,      deg,    (size_t)(N + 1) * 4);
    TAKE(int*,      offs,   (size_t)(N + 1) * 4);
    TAKE(int*,      cursor, (size_t)N * 4);
    TAKE(unsigned*, hg,     (size_t)NGRAPH * DF * 4);
#undef TAKE

    // ---- build dst-CSR once (reused for all 3 layers) ----
    zero_u32<<<(N + 1 + 255) / 256, 256, 0, stream>>>((unsigned*)deg, N + 1);
    count_deg<<<(E + 255) / 256, 256, 0, stream>>>(dst, deg, E);
    scan_excl<<<1, 1024, 0, stream>>>(deg, offs, N);
    zero_u32<<<(N + 255) / 256, 256, 0, stream>>>((unsigned*)cursor, N);
    fill_csr<<<(E + 255) / 256, 256, 0, stream>>>(src, dst, offs, cursor, csr, E);

    // ---- 3 EdgeConv layers ----
    const float* hcur = h;
    float* pbuf[2] = {p0, p1};
    for (int l = 0; l < 3; ++l) {
        float* pl = pbuf[l & 1];
        gemm_dual_wmma<<<(N + 127) / 128, 256, 0, stream>>>(
            hcur, thw + (size_t)l * DF * DF, phw + (size_t)l * DF * DF,
            thb + (size_t)l * DF, phb + (size_t)l * DF, t_buf, pl, N);
        edge_agg<<<(N + 7) / 8, 256, 0, stream>>>(t_buf, pl, offs, csr, N);
        hcur = pl;
    }

    // ---- readout ----
    zero_u32<<<(NGRAPH * DF + 255) / 256, 256, 0, stream>>>(hg, NGRAPH * DF);
    {
        int segs = (N + 63) / 64;
        pool_max<<<(segs + 3) / 4, 256, 0, stream>>>(hcur, gid, hg, N);
    }
    classify<<<1, 1024, 0, stream>>>(hg, clw, clb, (float*)d_out);
}